// Stage_47287589929774
// MI455X (gfx1250) — compile-verified
//
#include <hip/hip_runtime.h>
#include <hip/hip_bf16.h>

typedef __attribute__((ext_vector_type(2))) float v2f;
typedef __attribute__((ext_vector_type(8))) float v8f;

#define BN_EPS 1e-5f

// ---------------------------------------------------------------------------
// gfx1250 async copy: global -> LDS, per-lane addressed on BOTH sides
// (lane supplies its own global source and LDS destination), ASYNCcnt-tracked.
// ---------------------------------------------------------------------------
__device__ __forceinline__ void async_copy_b32(unsigned int lds_off,
                                               const void* gptr) {
  unsigned long long ga = (unsigned long long)(uintptr_t)gptr;
  asm volatile("global_load_async_to_lds_b32 %0, %1, off"
               :: "v"(lds_off), "v"(ga) : "memory");
}
__device__ __forceinline__ void wait_async0() {
  asm volatile("s_wait_asynccnt 0x0" ::: "memory");
}

// ---------------------------------------------------------------------------
// WMMA GEMM + fused BN batch-stat accumulation.
//   Y(RxC) = X(RxK) @ W(KxC) + bias,  gsum/gsum2 += per-channel sum / sum^2
// One wave -> one 16-row stripe across all C columns (CT = C/16 tiles);
// the A fragment is loaded once per k-step (global_load_b64) and reused for
// all CT WMMAs.  W is staged once per block into LDS in a pair-interleaved,
// bank-padded layout so every B fragment is a single aligned ds_load_b64:
//   element (k, col) -> lwp[(k>>1)*CP2 + 2*col + (k&1)],  CP2 = 2C+2
// V_WMMA_F32_16X16X4_F32 fragment layout (ISA 7.12.2):
//   A 16x4 : lanes 0-15 (M=lane, K={0,1}), lanes 16-31 (M=lane-16, K={2,3})
//   B 4x16 : lanes 0-15 (K={0,1}, N=lane), lanes 16-31 (K={2,3}, N=lane-16)
//   C/D    : VGPR j -> M=j (+8 for upper half-wave), N = lane&15
// ---------------------------------------------------------------------------
template<int K, int CT>
__global__ __launch_bounds__(256) void gemm_bias_stats_wmma(
    const float* __restrict__ X, const float* __restrict__ W,
    const float* __restrict__ bias, float* __restrict__ Y,
    float* __restrict__ gsum, float* __restrict__ gsum2, int R)
{
  constexpr int C   = CT * 16;
  constexpr int CP2 = 2 * C + 2;       // padded row-pair stride (floats)
  __shared__ float lwp[(K / 2) * CP2]; // pair-interleaved weights
  __shared__ float lsum[C], lsum2[C];  // block-level stat accumulators

  // ---- stage W: coalesced global reads, scattered async writes into LDS ----
  {
    unsigned int base = (unsigned int)(uintptr_t)lwp;   // LDS byte offset
    for (int i = threadIdx.x; i < K * C; i += 256) {
      int k = i / C, col = i & (C - 1);                 // i = k*C + col
      unsigned int dst = (unsigned int)(((k >> 1) * CP2 + 2 * col + (k & 1)) * 4);
      async_copy_b32(base + dst, (const char*)W + 4u * (unsigned)i);
    }
    if (threadIdx.x < C) { lsum[threadIdx.x] = 0.f; lsum2[threadIdx.x] = 0.f; }
    wait_async0();                                      // this wave's asyncs done
  }
  __syncthreads();                                      // whole block staged

  const int lane = threadIdx.x & 31;
  const int wave = threadIdx.x >> 5;
  const int rt   = blockIdx.x * 8 + wave;
  const int rtiles = (R + 15) >> 4;
  const int lm = lane & 15;
  const int hi = lane >> 4;

  if (rt < rtiles) {                                    // wave-uniform
    const int r0 = rt << 4;
    int rowA = r0 + lm; rowA = rowA < R ? rowA : R - 1; // clamp, no EXEC change
    const float* __restrict__ xr = X + (size_t)rowA * K + 2 * hi;

    v8f acc[CT];
    const v8f vzero = {};
#pragma unroll
    for (int t = 0; t < CT; ++t) acc[t] = vzero;

#pragma unroll 4
    for (int k = 0; k < K; k += 4) {
      v2f a;
      a.x = xr[k];
      a.y = xr[k + 1];
      // row pair index: rows (k+2*hi, k+2*hi+1) -> pair (k>>1)+hi
      const float* __restrict__ lwr = lwp + ((k >> 1) + hi) * CP2 + 2 * lm;
#pragma unroll
      for (int t = 0; t < CT; ++t) {
        v2f b = *(const v2f*)(lwr + 32 * t);            // one ds_load_b64
        acc[t] = __builtin_amdgcn_wmma_f32_16x16x4_f32(
            false, a, false, b, (short)0, acc[t], false, false);
      }
    }

    // ---- epilogue: bias, store, per-column stat partials ----
#pragma unroll
    for (int t = 0; t < CT; ++t) {
      const int col = t * 16 + lm;
      const float bv = bias[col];
      float s = 0.f, s2 = 0.f;
#pragma unroll
      for (int j = 0; j < 8; ++j) {
        int row = r0 + j + hi * 8;
        float v = acc[t][j] + bv;
        if (row < R) {
          Y[(size_t)row * C + col] = v;
          s += v; s2 += v * v;
        }
      }
      // lanes l and l+16 hold the same column (different row halves)
      s  += __shfl_xor(s, 16, 32);
      s2 += __shfl_xor(s2, 16, 32);
      if (lane < 16) {
        atomicAdd(&lsum[col], s);        // ds_add_f32
        atomicAdd(&lsum2[col], s2);
      }
    }
  }
  __syncthreads();
  if (threadIdx.x < C) {
    atomicAdd(&gsum[threadIdx.x],  lsum[threadIdx.x]);   // global_atomic_add_f32
    atomicAdd(&gsum2[threadIdx.x], lsum2[threadIdx.x]);
  }
}

// ---------------------------------------------------------------------------
// BN stat helpers
// ---------------------------------------------------------------------------
__global__ __launch_bounds__(256) void zero_stats(float* __restrict__ acc) {
  if (threadIdx.x < 256) acc[threadIdx.x] = 0.f;   // gsum[0:128], gsum2[128:256]
}

__global__ __launch_bounds__(128) void stats_finalize(
    const float* __restrict__ gsum, const float* __restrict__ gsum2,
    float* __restrict__ mean, float* __restrict__ inv, int R, int C)
{
  int c = threadIdx.x;
  if (c >= C) return;
  float m = gsum[c] / (float)R;
  float var = gsum2[c] / (float)R - m * m;   // biased var, as jnp.var
  mean[c] = m;
  inv[c]  = rsqrtf(var + BN_EPS);
}

// Fallback full-pass stats for non-GEMM producers (small arrays only)
__global__ __launch_bounds__(256) void col_stats(
    const float* __restrict__ Y, float* __restrict__ mean,
    float* __restrict__ inv, int R, int C)
{
  const int c = blockIdx.x;
  float s = 0.f, s2 = 0.f;
  for (int r = threadIdx.x; r < R; r += blockDim.x) {
    float v = Y[(size_t)r * C + c];
    s += v; s2 += v * v;
  }
  __shared__ float ls[256], ls2[256];
  ls[threadIdx.x] = s; ls2[threadIdx.x] = s2;
  __syncthreads();
  for (int o = 128; o > 0; o >>= 1) {
    if (threadIdx.x < o) {
      ls[threadIdx.x]  += ls[threadIdx.x + o];
      ls2[threadIdx.x] += ls2[threadIdx.x + o];
    }
    __syncthreads();
  }
  if (threadIdx.x == 0) {
    float m = ls[0] / (float)R;
    float var = ls2[0] / (float)R - m * m;
    mean[c] = m;
    inv[c]  = rsqrtf(var + BN_EPS);
  }
}

// In-place BN apply (+ optional ReLU); C is a power of two
__global__ __launch_bounds__(256) void bn_apply(
    float* __restrict__ Y, const float* __restrict__ mean,
    const float* __restrict__ inv, const float* __restrict__ g,
    const float* __restrict__ bt, size_t total, int Cmask, int relu)
{
  size_t i = (size_t)blockIdx.x * blockDim.x + threadIdx.x;
  if (i >= total) return;
  int c = (int)i & Cmask;
  float v = (Y[i] - mean[c]) * inv[c] * g[c] + bt[c];
  if (relu) v = fmaxf(v, 0.f);
  Y[i] = v;
}

// ---------------------------------------------------------------------------
// Misc elementwise / gather kernels
// ---------------------------------------------------------------------------
__global__ __launch_bounds__(256) void gather_points(
    const float* __restrict__ p, const int* __restrict__ index,
    float* __restrict__ out, int M)
{
  int i = blockIdx.x * blockDim.x + threadIdx.x;
  if (i >= M) return;
  int src = index[i];
  out[i * 3 + 0] = p[(size_t)src * 3 + 0];
  out[i * 3 + 1] = p[(size_t)src * 3 + 1];
  out[i * 3 + 2] = p[(size_t)src * 3 + 2];
}

// Layer 0 of mlp3: dp = psrc[knn] - center, then dp @ W0(3x32) + b0
__global__ __launch_bounds__(256) void dp_layer0(
    const float* __restrict__ psrc, const float* __restrict__ center,
    const int* __restrict__ knn, const float* __restrict__ W0,
    const float* __restrict__ b0, float* __restrict__ T0, int RNS, int NS)
{
  int row = blockIdx.x * blockDim.x + threadIdx.x;
  if (row >= RNS) return;
  int m = row / NS;
  int idx = knn[row];
  float dx = psrc[(size_t)idx * 3 + 0] - center[(size_t)m * 3 + 0];
  float dy = psrc[(size_t)idx * 3 + 1] - center[(size_t)m * 3 + 1];
  float dz = psrc[(size_t)idx * 3 + 2] - center[(size_t)m * 3 + 2];
  float* __restrict__ o = T0 + (size_t)row * 32;
#pragma unroll
  for (int c = 0; c < 32; ++c)
    o[c] = fmaf(dx, W0[c], fmaf(dy, W0[32 + c], fmaf(dz, W0[64 + c], b0[c])));
}

// out[m,c] = max_s ( pe[(m*NS+s), c] + feat[knn[m*NS+s], c] )
__global__ __launch_bounds__(128) void pool_max(
    const float* __restrict__ pe, const float* __restrict__ feat,
    const int* __restrict__ knn, float* __restrict__ out, int NS, int C)
{
  int m = blockIdx.x;
  int c = threadIdx.x;
  float best = -3.4e38f;
  for (int s = 0; s < NS; ++s) {
    int row = m * NS + s;
    int idx = knn[row];
    float v = pe[(size_t)row * C + c] + feat[(size_t)idx * C + c];
    best = fmaxf(best, v);
  }
  out[(size_t)m * C + c] = best;
}

__global__ __launch_bounds__(256) void residual_relu(
    float* __restrict__ nf, const float* __restrict__ h, size_t total)
{
  size_t i = (size_t)blockIdx.x * blockDim.x + threadIdx.x;
  if (i < total) nf[i] = fmaxf(nf[i] + h[i], 0.f);
}

__global__ __launch_bounds__(256) void copy_out(
    const float* __restrict__ src, float* __restrict__ dst, size_t total)
{
  size_t i = (size_t)blockIdx.x * blockDim.x + threadIdx.x;
  if (i < total) dst[i] = src[i];
}

// ---------------------------------------------------------------------------
// Host-side helpers.  STAT layout: mean[0:128] inv[128:256] gsum[256:384]
// gsum2[384:512].
// ---------------------------------------------------------------------------
static inline void launch_gemm_bn(const float* X, const float* W, const float* b,
                                  const float* g, const float* bt, float* Y,
                                  float* STAT, int R, int K, int C, int relu,
                                  hipStream_t st)
{
  float* mean = STAT, *inv = STAT + 128, *gs = STAT + 256, *gs2 = STAT + 384;
  zero_stats<<<1, 256, 0, st>>>(gs);
  int rtiles = (R + 15) >> 4;
  int blocks = (rtiles + 7) / 8;                 // 8 waves / block
  if (K == 64)        gemm_bias_stats_wmma<64, 8><<<blocks, 256, 0, st>>>(X, W, b, Y, gs, gs2, R);
  else if (K == 128)  gemm_bias_stats_wmma<128, 8><<<blocks, 256, 0, st>>>(X, W, b, Y, gs, gs2, R);
  else if (C == 32)   gemm_bias_stats_wmma<32, 2><<<blocks, 256, 0, st>>>(X, W, b, Y, gs, gs2, R);
  else                gemm_bias_stats_wmma<32, 8><<<blocks, 256, 0, st>>>(X, W, b, Y, gs, gs2, R);
  stats_finalize<<<1, 128, 0, st>>>(gs, gs2, mean, inv, R, C);
  size_t tot = (size_t)R * C;
  bn_apply<<<(int)((tot + 255) / 256), 256, 0, st>>>(Y, mean, inv, g, bt, tot, C - 1, relu);
}

static inline void bn_sa(float* Y, const float* g, const float* bt, float* STAT,
                         int R, int C, int relu, hipStream_t st)
{
  col_stats<<<C, 256, 0, st>>>(Y, STAT, STAT + 128, R, C);
  size_t tot = (size_t)R * C;
  bn_apply<<<(int)((tot + 255) / 256), 256, 0, st>>>(Y, STAT, STAT + 128, g, bt, tot, C - 1, relu);
}

// Full 3->32->32->128 mlp3 with per-layer batch-stat BN (ReLU on first two)
static inline void run_mlp3(const float* psrc, const float* center, const int* knn,
                            void* const* d_in, int base,
                            float* T0, float* T1, float* OUT, float* STAT,
                            int M, int NS, hipStream_t st)
{
  int RNS = M * NS;
  dp_layer0<<<(RNS + 255) / 256, 256, 0, st>>>(
      psrc, center, knn, (const float*)d_in[base], (const float*)d_in[base + 1], T0, RNS, NS);
  bn_sa(T0, (const float*)d_in[base + 2], (const float*)d_in[base + 3], STAT, RNS, 32, 1, st);
  launch_gemm_bn(T0, (const float*)d_in[base + 4], (const float*)d_in[base + 5],
                 (const float*)d_in[base + 6], (const float*)d_in[base + 7],
                 T1, STAT, RNS, 32, 32, 1, st);
  launch_gemm_bn(T1, (const float*)d_in[base + 8], (const float*)d_in[base + 9],
                 (const float*)d_in[base + 10], (const float*)d_in[base + 11],
                 OUT, STAT, RNS, 32, 128, 0, st);
}

// ---------------------------------------------------------------------------
// Leaf order of d_in (pytree flatten of setup_inputs dict):
//  0 p, 1 f, 2-5 conv1 {W,b,g,bt}, 6-17 conv2 3x{W,b,g,bt}, 18-19 ds_bn {g,b},
//  20-31 pe_params 3x{W,b,g,bt}, 32-73 dsa 3x{preW,preb,preg,prebt, bng,bnb,
//  f0W,f0b,f0g,f0bt, f1W,f1b,f1g,f1bt}, 74 index, 75 knn_down, 76 knn_idx
// Workspace requirement: ~410 MB of float scratch.
// ---------------------------------------------------------------------------
extern "C" void kernel_launch(void* const* d_in, const int* in_sizes, int n_in,
                              void* d_out, int out_size, void* d_ws, size_t ws_size,
                              hipStream_t stream)
{
  (void)n_in; (void)out_size; (void)ws_size;
  const float* p        = (const float*)d_in[0];
  const float* f        = (const float*)d_in[1];
  const int*   index    = (const int*)d_in[74];
  const int*   knn_down = (const int*)d_in[75];
  const int*   knn_idx  = (const int*)d_in[76];

  const int N   = in_sizes[0] / 3;     // 100000
  const int M   = in_sizes[74];        // 25000
  const int NS  = 16;
  const int CIN = 64;
  const int C   = 128;
  const int RNS = M * NS;

  float* newp = (float*)d_out;                    // (M,3)
  float* outf = (float*)d_out + (size_t)M * 3;    // (M,128)

  float* w = (float*)d_ws;
  size_t off = 0;
  float* F1   = w + off; off += (size_t)N * C;      // post-BN-ReLU f1
  float* T0   = w + off; off += (size_t)RNS * 32;
  float* T1   = w + off; off += (size_t)RNS * 32;
  float* BIG  = w + off; off += (size_t)RNS * C;    // pe_d, then reused for pe
  float* NF   = w + off; off += (size_t)M * C;
  float* H    = w + off; off += (size_t)M * C;
  float* H2   = w + off; off += (size_t)M * C;
  float* H3   = w + off; off += (size_t)M * C;
  float* STAT = w + off; off += 512;

  // ---- DownSample ----
  gather_points<<<(M + 255) / 256, 256, 0, stream>>>(p, index, newp, M);

  // f1 = lin_bn(f, conv1)
  launch_gemm_bn(f, (const float*)d_in[2], (const float*)d_in[3],
                 (const float*)d_in[4], (const float*)d_in[5],
                 F1, STAT, N, CIN, C, 1, stream);

  // pe_d = mlp3(p[knn_down] - new_p, conv2)  -> BIG
  run_mlp3(p, newp, knn_down, d_in, 6, T0, T1, BIG, STAT, M, NS, stream);

  // nf = bn(max_s(pe_d + f1[knn_down]), ds_bn)
  pool_max<<<M, C, 0, stream>>>(BIG, F1, knn_down, NF, NS, C);
  bn_sa(NF, (const float*)d_in[18], (const float*)d_in[19], STAT, M, C, 0, stream);

  // pe = mlp3(new_p[knn_idx] - new_p, pe_params) -> BIG (pe_d consumed)
  run_mlp3(newp, newp, knn_idx, d_in, 20, T0, T1, BIG, STAT, M, NS, stream);

  // ---- DSA residual blocks ----
  const size_t totMC = (size_t)M * C;
  const int blkMC = (int)((totMC + 255) / 256);
  for (int i = 0; i < 3; ++i) {
    int b = 32 + 14 * i;
    // h = lin_bn(nf, pre)
    launch_gemm_bn(NF, (const float*)d_in[b], (const float*)d_in[b + 1],
                   (const float*)d_in[b + 2], (const float*)d_in[b + 3],
                   H, STAT, M, C, C, 1, stream);
    // h = bn(max_s(pe + h[knn_idx]), bnp)
    pool_max<<<M, C, 0, stream>>>(BIG, H, knn_idx, H2, NS, C);
    bn_sa(H2, (const float*)d_in[b + 4], (const float*)d_in[b + 5], STAT, M, C, 0, stream);
    // h = lin_bn(h, ffn0)
    launch_gemm_bn(H2, (const float*)d_in[b + 6], (const float*)d_in[b + 7],
                   (const float*)d_in[b + 8], (const float*)d_in[b + 9],
                   H3, STAT, M, C, C, 1, stream);
    // h = lin_bn(h, ffn1, act=False)
    launch_gemm_bn(H3, (const float*)d_in[b + 10], (const float*)d_in[b + 11],
                   (const float*)d_in[b + 12], (const float*)d_in[b + 13],
                   H, STAT, M, C, C, 0, stream);
    // nf = relu(identity + h)
    residual_relu<<<blkMC, 256, 0, stream>>>(NF, H, totMC);
  }

  copy_out<<<blkMC, 256, 0, stream>>>(NF, outf, totMC);
}